// GCNFeatures_88244398064244
// MI455X (gfx1250) — compile-verified
//
#include <hip/hip_runtime.h>
#include <hip/hip_bf16.h>

typedef float v2f __attribute__((ext_vector_type(2)));
typedef float v8f __attribute__((ext_vector_type(8)));

// ---------------------------------------------------------------------------
// utility fills
// ---------------------------------------------------------------------------
__global__ void k_fill(float* __restrict__ p, long long n, float v) {
  long long i = (long long)blockIdx.x * blockDim.x + threadIdx.x;
  if (i < n) p[i] = v;
}

// deg[dst] += 1 per edge (deg pre-filled with 1.0 for the self loop)
__global__ void k_degree(const int* __restrict__ dst, int e, float* __restrict__ deg) {
  int i = blockIdx.x * blockDim.x + threadIdx.x;
  if (i < e) unsafeAtomicAdd(&deg[dst[i]], 1.0f);
}

__global__ void k_rsqrt(float* __restrict__ d, int n) {
  int i = blockIdx.x * blockDim.x + threadIdx.x;
  if (i < n) d[i] = rsqrtf(d[i]);
}

// ---------------------------------------------------------------------------
// H = X * W  (X:[n,64], W:[64,64]) using V_WMMA_F32_16X16X4_F32.
// One block = 128 threads = 4 waves; block handles one 16-row tile, each wave
// one 16-col tile. W is staged in LDS once per block.
//
// f32 A-frag layout (16x4):  lanes 0-15 -> K={kb,kb+1}, lanes 16-31 -> K={kb+2,kb+3}
// f32 B-frag layout (4x16):  same K split, N = lane%16
// f32 C/D layout: VGPR r -> M=r (lanes 0-15), M=r+8 (lanes 16-31), N=lane%16
// ---------------------------------------------------------------------------
__global__ void __launch_bounds__(128)
k_gemm_wmma(const float* __restrict__ X, const float* __restrict__ W,
            float* __restrict__ H, int n) {
  __shared__ float Ws[64 * 64];
  for (int i = threadIdx.x; i < 64 * 16; i += 128)
    ((float4*)Ws)[i] = ((const float4*)W)[i];
  __syncthreads();

  const int tile  = blockIdx.x;          // 16-row tile index
  const int ntile = threadIdx.x >> 5;    // wave id = output col tile (0..3)
  const int lane  = threadIdx.x & 31;
  const int m     = lane & 15;
  const int hi    = lane >> 4;           // 0 -> K pair {0,1}, 1 -> {2,3}
  const int col   = (ntile << 4) + m;

  int row = tile * 16 + m;
  if (row >= n) row = n - 1;             // clamp: WMMA needs full EXEC
  const float* xrow = X + (long long)row * 64;

  v8f acc = {0.f, 0.f, 0.f, 0.f, 0.f, 0.f, 0.f, 0.f};
#pragma unroll
  for (int kb = 0; kb < 64; kb += 4) {
    const int k0 = kb + (hi << 1);
    v2f a, b;
    a.x = xrow[k0];
    a.y = xrow[k0 + 1];
    b.x = Ws[k0 * 64 + col];
    b.y = Ws[(k0 + 1) * 64 + col];
    acc = __builtin_amdgcn_wmma_f32_16x16x4_f32(false, a, false, b,
                                                (short)0, acc, false, false);
  }

  const int rbase = tile * 16 + (hi << 3);
#pragma unroll
  for (int r = 0; r < 8; ++r) {
    const int grow = rbase + r;
    if (grow < n) H[(long long)grow * 64 + col] = acc[r];
  }
}

// ---------------------------------------------------------------------------
// agg[dst] += h[src] * dinv[src]*dinv[dst] ; 16 threads per edge, float4 each
// ---------------------------------------------------------------------------
__global__ void __launch_bounds__(256)
k_scatter(const float* __restrict__ H, const int* __restrict__ src,
          const int* __restrict__ dst, const float* __restrict__ dinv,
          float* __restrict__ agg, int e) {
  long long idx = (long long)blockIdx.x * blockDim.x + threadIdx.x;
  long long tot = (long long)e * 16;
  if (idx >= tot) return;
  const int ei = (int)(idx >> 4);
  const int c4 = (int)(idx & 15) << 2;
  const int s = src[ei];
  const int d = dst[ei];
  const float coef = dinv[s] * dinv[d];
  const float4 hv = *(const float4*)(H + (long long)s * 64 + c4);
  float* ap = agg + (long long)d * 64 + c4;
  unsafeAtomicAdd(ap + 0, hv.x * coef);
  unsafeAtomicAdd(ap + 1, hv.y * coef);
  unsafeAtomicAdd(ap + 2, hv.z * coef);
  unsafeAtomicAdd(ap + 3, hv.w * coef);
}

// ---------------------------------------------------------------------------
// out = relu(agg + h*dinv^2 + b)   (out may alias agg: pure elementwise)
// ---------------------------------------------------------------------------
__global__ void __launch_bounds__(256)
k_post(const float* __restrict__ agg, const float* __restrict__ H,
       const float* __restrict__ dinv, const float* __restrict__ b,
       float* __restrict__ out, int n) {
  long long idx = (long long)blockIdx.x * blockDim.x + threadIdx.x;
  long long tot = (long long)n * 64;
  if (idx >= tot) return;
  const int i = (int)(idx >> 6);
  const int c = (int)(idx & 63);
  const float di = dinv[i];
  const float v = agg[idx] + H[idx] * di * di + b[c];
  out[idx] = fmaxf(v, 0.0f);
}

// ---------------------------------------------------------------------------
// y = softmax(x @ fcW + fcb), fcW:[64,8]; one thread per node, fcW in LDS
// ---------------------------------------------------------------------------
__global__ void __launch_bounds__(256)
k_fc_softmax(const float* __restrict__ X, const float* __restrict__ fcW,
             const float* __restrict__ fcb, float* __restrict__ Y, int n) {
  __shared__ float Wf[64 * 8];
  __shared__ float Bf[8];
  for (int i = threadIdx.x; i < 64 * 8; i += 256) Wf[i] = fcW[i];
  if (threadIdx.x < 8) Bf[threadIdx.x] = fcb[threadIdx.x];
  __syncthreads();

  const int i = blockIdx.x * blockDim.x + threadIdx.x;
  if (i >= n) return;
  const float* xr = X + (long long)i * 64;

  float acc[8];
#pragma unroll
  for (int c = 0; c < 8; ++c) acc[c] = Bf[c];
  for (int k = 0; k < 64; ++k) {
    const float xv = xr[k];
#pragma unroll
    for (int c = 0; c < 8; ++c) acc[c] = fmaf(xv, Wf[k * 8 + c], acc[c]);
  }
  float mx = acc[0];
#pragma unroll
  for (int c = 1; c < 8; ++c) mx = fmaxf(mx, acc[c]);
  float s = 0.0f;
#pragma unroll
  for (int c = 0; c < 8; ++c) { acc[c] = __expf(acc[c] - mx); s += acc[c]; }
  const float inv = 1.0f / s;
#pragma unroll
  for (int c = 0; c < 8; ++c) Y[(long long)i * 8 + c] = acc[c] * inv;
}

// ---------------------------------------------------------------------------
extern "C" void kernel_launch(void* const* d_in, const int* in_sizes, int n_in,
                              void* d_out, int out_size, void* d_ws, size_t ws_size,
                              hipStream_t stream) {
  (void)n_in; (void)out_size; (void)ws_size;

  const float* x0  = (const float*)d_in[0];
  const int*   ei  = (const int*)  d_in[1];
  const float* W1  = (const float*)d_in[2];
  const float* b1  = (const float*)d_in[3];
  const float* W2  = (const float*)d_in[4];
  const float* b2  = (const float*)d_in[5];
  const float* W3  = (const float*)d_in[6];
  const float* b3  = (const float*)d_in[7];
  const float* fcW = (const float*)d_in[8];
  const float* fcb = (const float*)d_in[9];

  const int n = in_sizes[0] / 64;   // 100000
  const int e = in_sizes[1] / 2;    // 1250000
  const int* src = ei;
  const int* dst = ei + e;

  float* ws   = (float*)d_ws;
  float* H    = ws;                              // [n*64]
  float* aggA = H    + (long long)n * 64;        // [n*64]
  float* aggB = aggA + (long long)n * 64;        // [n*64]
  float* dinv = aggB + (long long)n * 64;        // [n]

  float* outx = (float*)d_out;                   // [n*64]
  float* outy = outx + (long long)n * 64;        // [n*8]

  const long long nd   = (long long)n * 64;
  const int blk = 256;
  const int gN   = (int)((n + blk - 1) / blk);
  const int gND  = (int)((nd + blk - 1) / blk);
  const int gE   = (int)((e + blk - 1) / blk);
  const int gE16 = (int)(((long long)e * 16 + blk - 1) / blk);
  const int gT   = (n + 15) / 16;                // 16-row GEMM tiles

  // degrees -> dinv
  k_fill<<<gN, blk, 0, stream>>>(dinv, n, 1.0f);
  k_degree<<<gE, blk, 0, stream>>>(dst, e, dinv);
  k_rsqrt<<<gN, blk, 0, stream>>>(dinv, n);

  // layer 1: x0 -> aggA
  k_fill<<<gND, blk, 0, stream>>>(aggA, nd, 0.0f);
  k_gemm_wmma<<<gT, 128, 0, stream>>>(x0, W1, H, n);
  k_scatter<<<gE16, blk, 0, stream>>>(H, src, dst, dinv, aggA, e);
  k_post<<<gND, blk, 0, stream>>>(aggA, H, dinv, b1, aggA, n);

  // layer 2: aggA -> aggB
  k_fill<<<gND, blk, 0, stream>>>(aggB, nd, 0.0f);
  k_gemm_wmma<<<gT, 128, 0, stream>>>(aggA, W2, H, n);
  k_scatter<<<gE16, blk, 0, stream>>>(H, src, dst, dinv, aggB, e);
  k_post<<<gND, blk, 0, stream>>>(aggB, H, dinv, b2, aggB, n);

  // layer 3: aggB -> d_out (x part), reusing aggA as accumulator
  k_fill<<<gND, blk, 0, stream>>>(aggA, nd, 0.0f);
  k_gemm_wmma<<<gT, 128, 0, stream>>>(aggB, W3, H, n);
  k_scatter<<<gE16, blk, 0, stream>>>(H, src, dst, dinv, aggA, e);
  k_post<<<gND, blk, 0, stream>>>(aggA, H, dinv, b3, outx, n);

  // FC + softmax -> d_out (y part)
  k_fc_softmax<<<gN, blk, 0, stream>>>(outx, fcW, fcb, outy, n);
}